// GraphSAGEBaseline_83545703841880
// MI455X (gfx1250) — compile-verified
//
#include <hip/hip_runtime.h>
#include <math.h>

#define N_NODES 100000
#define N_EDGES 1600000
#define DIM     128
#define EQN     262144

typedef __attribute__((ext_vector_type(2))) float v2f;
typedef __attribute__((ext_vector_type(8))) float v8f;

__device__ __forceinline__ v8f wmma_f32_4(v2f a, v2f b, v8f c) {
  // V_WMMA_F32_16X16X4_F32 : D = A(16x4 f32) x B(4x16 f32) + C(16x16 f32)
  return __builtin_amdgcn_wmma_f32_16x16x4_f32(
      /*neg_a=*/false, a, /*neg_b=*/false, b,
      /*c_mod=*/(short)0, c, /*reuse_a=*/false, /*reuse_b=*/false);
}

// ---------------------------------------------------------------------------
// Zero-fill (agg + cnt must be zeroed deterministically every call)
// ---------------------------------------------------------------------------
__global__ void zero_kernel(float* __restrict__ p, int n) {
  for (int i = blockIdx.x * blockDim.x + threadIdx.x; i < n;
       i += gridDim.x * blockDim.x)
    p[i] = 0.0f;
}

// ---------------------------------------------------------------------------
// Scatter-sum + degree count. One wave per edge: 32 lanes x float4 = 128 f32.
// HW f32 atomics (global_atomic_add_f32) — this kernel is the HBM roofline.
// ---------------------------------------------------------------------------
__global__ __launch_bounds__(256)
void scatter_kernel(const float* __restrict__ x, const int* __restrict__ ei,
                    float* __restrict__ agg, float* __restrict__ cnt) {
  const int wave = threadIdx.x >> 5;
  const int lane = threadIdx.x & 31;
  const int e = blockIdx.x * 8 + wave;
  if (e >= N_EDGES) return;
  const int src = ei[e];
  const int dst = ei[N_EDGES + e];
  const float4 v =
      reinterpret_cast<const float4*>(x + (size_t)src * DIM)[lane];
  float* ap = agg + (size_t)dst * DIM + lane * 4;
  unsafeAtomicAdd(ap + 0, v.x);
  unsafeAtomicAdd(ap + 1, v.y);
  unsafeAtomicAdd(ap + 2, v.z);
  unsafeAtomicAdd(ap + 3, v.w);
  if (lane == 0) unsafeAtomicAdd(cnt + dst, 1.0f);
}

// ---------------------------------------------------------------------------
// SAGE layer: y = relu(l2norm(mean @ Wl^T + bl + x @ Wr^T))
// Block = 256 threads (8 waves) computes a 16x128 output row-block.
// Wave w owns columns [16w, 16w+16); K-loop of 32 f32 WMMAs per GEMM.
// ---------------------------------------------------------------------------
__global__ __launch_bounds__(256)
void sage_kernel(const float* __restrict__ agg, const float* __restrict__ cnt,
                 const float* __restrict__ x, const float* __restrict__ Wl,
                 const float* __restrict__ bl, const float* __restrict__ Wr,
                 float* __restrict__ y) {
  __shared__ float Atile[16 * DIM];  // mean rows
  __shared__ float Xtile[16 * DIM];  // root rows
  __shared__ float rowsq[16];

  const int tid = threadIdx.x;
  const int r0 = blockIdx.x * 16;

  if (tid < 16) rowsq[tid] = 0.0f;
  for (int idx = tid; idx < 16 * DIM; idx += 256) {
    const int r = idx >> 7, c = idx & (DIM - 1);
    const int g = r0 + r;
    const float inv = 1.0f / fmaxf(cnt[g], 1.0f);
    Atile[idx] = agg[(size_t)g * DIM + c] * inv;
    Xtile[idx] = x[(size_t)g * DIM + c];
  }
  __syncthreads();

  const int wave = tid >> 5, lane = tid & 31;
  const int mrow = lane & 15;          // A fragment row
  const int kb   = (lane >> 4) * 2;    // K pair selected by lane half
  const int col  = wave * 16 + (lane & 15);

  v8f acc = {};
  // mean @ Wl^T   (B[k][col] = Wl[col*DIM + k])
  const float* BW = Wl + (size_t)col * DIM;
#pragma unroll
  for (int k0 = 0; k0 < DIM; k0 += 4) {
    v2f a, b;
    a.x = Atile[mrow * DIM + k0 + kb];
    a.y = Atile[mrow * DIM + k0 + kb + 1];
    b.x = BW[k0 + kb];
    b.y = BW[k0 + kb + 1];
    acc = wmma_f32_4(a, b, acc);
  }
  // + x @ Wr^T
  const float* BW2 = Wr + (size_t)col * DIM;
#pragma unroll
  for (int k0 = 0; k0 < DIM; k0 += 4) {
    v2f a, b;
    a.x = Xtile[mrow * DIM + k0 + kb];
    a.y = Xtile[mrow * DIM + k0 + kb + 1];
    b.x = BW2[k0 + kb];
    b.y = BW2[k0 + kb + 1];
    acc = wmma_f32_4(a, b, acc);
  }

  // bias + per-row sum of squares (reduce 16 columns within a lane-half)
  const float bias = bl[col];
  float sq[8];
#pragma unroll
  for (int i = 0; i < 8; ++i) {
    acc[i] += bias;
    float s = acc[i] * acc[i];
    s += __shfl_xor(s, 1, 32);
    s += __shfl_xor(s, 2, 32);
    s += __shfl_xor(s, 4, 32);
    s += __shfl_xor(s, 8, 32);
    sq[i] = s;
  }
  if ((lane & 15) == 0) {
    const int rb = (lane >> 4) * 8;
#pragma unroll
    for (int i = 0; i < 8; ++i) atomicAdd(&rowsq[rb + i], sq[i]);  // ds_add_f32
  }
  __syncthreads();

  const int rb = (lane >> 4) * 8;  // C/D layout: VGPR i -> M = rb + i
#pragma unroll
  for (int i = 0; i < 8; ++i) {
    const float nrm = sqrtf(rowsq[rb + i]);
    const float inv = 1.0f / fmaxf(nrm, 1e-12f);
    y[(size_t)(r0 + rb + i) * DIM + col] = fmaxf(acc[i] * inv, 0.0f);
  }
}

// ---------------------------------------------------------------------------
// Link predictor: per-block tile of 16 query edges.
//   e  = concat(x[row], x[col])                [16,256] in LDS
//   h1 = relu(e @ W1^T + b1)                   [16,128] (8 waves, 64 WMMAs)
//   h2 = relu(h1 @ W2^T + b2)                  [16, 64] (4 waves, 32 WMMAs)
//   out= sigmoid(h2 . w3 + b3)                 [16]
// ---------------------------------------------------------------------------
__global__ __launch_bounds__(256)
void mlp_kernel(const float* __restrict__ x, const int* __restrict__ eq,
                const float* __restrict__ W1, const float* __restrict__ b1,
                const float* __restrict__ W2, const float* __restrict__ b2,
                const float* __restrict__ W3, const float* __restrict__ b3,
                float* __restrict__ out) {
  __shared__ float Et[16 * 256];
  __shared__ float H1[16 * 128];
  __shared__ float H2[16 * 64];

  const int tid = threadIdx.x;
  const int e0 = blockIdx.x * 16;

  for (int idx = tid; idx < 16 * 256; idx += 256) {
    const int r = idx >> 8, c = idx & 255;
    const int e = e0 + r;
    const int node = (c < 128) ? eq[e] : eq[EQN + e];
    Et[idx] = x[(size_t)node * DIM + (c & 127)];
  }
  __syncthreads();

  const int wave = tid >> 5, lane = tid & 31;
  const int mrow = lane & 15;
  const int kb = (lane >> 4) * 2;
  const int rb = (lane >> 4) * 8;

  {  // layer 1: 256 -> 128
    const int col = wave * 16 + (lane & 15);
    v8f acc = {};
    const float* BW = W1 + (size_t)col * 256;
#pragma unroll
    for (int k0 = 0; k0 < 256; k0 += 4) {
      v2f a, b;
      a.x = Et[mrow * 256 + k0 + kb];
      a.y = Et[mrow * 256 + k0 + kb + 1];
      b.x = BW[k0 + kb];
      b.y = BW[k0 + kb + 1];
      acc = wmma_f32_4(a, b, acc);
    }
    const float bias = b1[col];
#pragma unroll
    for (int i = 0; i < 8; ++i)
      H1[(rb + i) * 128 + col] = fmaxf(acc[i] + bias, 0.0f);
  }
  __syncthreads();

  if (wave < 4) {  // layer 2: 128 -> 64 (whole waves, EXEC all-1 inside)
    const int col = wave * 16 + (lane & 15);
    v8f acc = {};
    const float* BW = W2 + (size_t)col * 128;
#pragma unroll
    for (int k0 = 0; k0 < 128; k0 += 4) {
      v2f a, b;
      a.x = H1[mrow * 128 + k0 + kb];
      a.y = H1[mrow * 128 + k0 + kb + 1];
      b.x = BW[k0 + kb];
      b.y = BW[k0 + kb + 1];
      acc = wmma_f32_4(a, b, acc);
    }
    const float bias = b2[col];
#pragma unroll
    for (int i = 0; i < 8; ++i)
      H2[(rb + i) * 64 + col] = fmaxf(acc[i] + bias, 0.0f);
  }
  __syncthreads();

  if (tid < 16) {  // layer 3: 64 -> 1 + sigmoid
    float z = b3[0];
#pragma unroll 8
    for (int k = 0; k < 64; ++k) z += H2[tid * 64 + k] * W3[k];
    out[e0 + tid] = 1.0f / (1.0f + expf(-z));
  }
}

// ---------------------------------------------------------------------------
extern "C" void kernel_launch(void* const* d_in, const int* in_sizes, int n_in,
                              void* d_out, int out_size, void* d_ws,
                              size_t ws_size, hipStream_t stream) {
  (void)in_sizes; (void)n_in; (void)out_size; (void)ws_size;

  const float* x   = (const float*)d_in[0];
  const int* ei    = (const int*)d_in[1];
  const int* eq    = (const int*)d_in[2];
  const float* Wl[3] = {(const float*)d_in[3], (const float*)d_in[6],
                        (const float*)d_in[9]};
  const float* bb[3] = {(const float*)d_in[4], (const float*)d_in[7],
                        (const float*)d_in[10]};
  const float* Wr[3] = {(const float*)d_in[5], (const float*)d_in[8],
                        (const float*)d_in[11]};
  const float* lpW1 = (const float*)d_in[12];
  const float* lpb1 = (const float*)d_in[13];
  const float* lpW2 = (const float*)d_in[14];
  const float* lpb2 = (const float*)d_in[15];
  const float* lpW3 = (const float*)d_in[16];
  const float* lpb3 = (const float*)d_in[17];

  // workspace layout: agg [N*D] | cnt [N] | h0 [N*D] | h1 [N*D]
  float* agg = (float*)d_ws;
  float* cnt = agg + (size_t)N_NODES * DIM;
  float* h0  = cnt + N_NODES;
  float* h1  = h0 + (size_t)N_NODES * DIM;

  const int zn = N_NODES * DIM + N_NODES;  // agg+cnt contiguous
  auto layer = [&](const float* xin, int i, float* yout) {
    zero_kernel<<<4096, 256, 0, stream>>>(agg, zn);
    scatter_kernel<<<(N_EDGES + 7) / 8, 256, 0, stream>>>(xin, ei, agg, cnt);
    sage_kernel<<<N_NODES / 16, 256, 0, stream>>>(agg, cnt, xin, Wl[i], bb[i],
                                                  Wr[i], yout);
  };
  layer(x, 0, h0);
  layer(h0, 1, h1);
  layer(h1, 2, h0);

  mlp_kernel<<<EQN / 16, 256, 0, stream>>>(h0, eq, lpW1, lpb1, lpW2, lpb2,
                                           lpW3, lpb3, (float*)d_out);
}